// GraphSage_29162827940511
// MI455X (gfx1250) — compile-verified
//
#include <hip/hip_runtime.h>
#include <hip/hip_bf16.h>
#include <math.h>

// GraphSAGE forward on MI455X (gfx1250, wave32).
// GEMMs use V_WMMA_F32_16X16X4_F32 (fp32 A/B -> fp32 acc, exact precision class).
// Gathers + 25-way max + concat are fused into the output GEMM's A-tile build.

typedef __attribute__((ext_vector_type(2))) float v2f;
typedef __attribute__((ext_vector_type(8))) float v8f;

#define N0 100000
#define N1 60000
#define N2 30000
#define SAMP 25
#define DIM 128

#define AS 132   // LDS row stride (floats) for 16x128 A tile: 132 % 64 == 4 -> conflict-free b64
#define CS 260   // LDS row stride (floats) for 16x256 cat tile: 260 % 64 == 4

#define STATS_BLOCKS 120

__device__ __forceinline__ float4 f4max(float4 a, float4 b) {
    float4 r;
    r.x = fmaxf(a.x, b.x); r.y = fmaxf(a.y, b.y);
    r.z = fmaxf(a.z, b.z); r.w = fmaxf(a.w, b.w);
    return r;
}

// --------------------------------------------------------------------------
// H = relu(X @ W^T + bias), X:[nrows,128], W:[128,128] row-major, H:[nrows,128]
// 256 threads = 8 waves; wave w owns output columns [16w,16w+16).
// B operand preloaded to 64 VGPRs/lane once; grid-stride over 16-row tiles.
// --------------------------------------------------------------------------
__global__ void __launch_bounds__(256) gemm_nt_relu(
    const float* __restrict__ X, const float* __restrict__ W,
    const float* __restrict__ bias, float* __restrict__ H, int nrows)
{
    __shared__ float As[16 * AS];
    const int wave = threadIdx.x >> 5;
    const int lane = threadIdx.x & 31;
    const int half = lane >> 4;       // 0: rows m..m / K 0,1 ; 1: K 2,3 / rows +8
    const int l16  = lane & 15;
    const int koff = half * 2;
    const int col  = wave * 16 + l16; // global output column for this lane

    // Preload B: b[s] = { W[col][4s+koff], W[col][4s+koff+1] }  (K step = 4)
    v2f breg[32];
#pragma unroll
    for (int s = 0; s < 32; ++s)
        breg[s] = *(const v2f*)(W + (size_t)col * DIM + s * 4 + koff);
    const float bcol = bias[col];

    const int ntiles = nrows >> 4;
    for (int tile = blockIdx.x; tile < ntiles; tile += gridDim.x) {
        const float* src = X + (size_t)tile * 16 * DIM;
        // cooperative 16x128 A-tile load (float4, fully coalesced)
        for (int i = threadIdx.x; i < 16 * 32; i += 256) {
            int r = i >> 5, cc = (i & 31) * 4;
            *(float4*)&As[r * AS + cc] = *(const float4*)(src + r * DIM + cc);
        }
        __syncthreads();

        v8f acc = {};
#pragma unroll
        for (int s = 0; s < 32; ++s) {
            v2f av = *(const v2f*)&As[l16 * AS + s * 4 + koff];
            acc = __builtin_amdgcn_wmma_f32_16x16x4_f32(
                false, av, false, breg[s], (short)0, acc, false, false);
        }

        // C/D layout: VGPR r -> row (r + 8*half) of the tile, column = col
        float* dst = H + (size_t)tile * 16 * DIM;
#pragma unroll
        for (int r = 0; r < 8; ++r) {
            float v = acc[r] + bcol;
            v = v > 0.f ? v : 0.f;
            dst[(r + half * 8) * DIM + col] = v;
        }
        __syncthreads();
    }
}

// --------------------------------------------------------------------------
// Fused SAGE layer output:
//   cat[m] = [ Xself[self_idx[m]] , max_{s<25} Hn[neigh_idx[m][s]] ]   (1x256)
//   Out[m] = cat[m] @ W^T + bias   (W:[128,256]);  optional relu.
// 256 threads = 8 waves; grid-stride over 16-node tiles.
// --------------------------------------------------------------------------
__global__ void __launch_bounds__(256) sage_fused(
    const float* __restrict__ Xself, const float* __restrict__ Hn,
    const int* __restrict__ neigh, const int* __restrict__ selfi,
    const float* __restrict__ W, const float* __restrict__ bias,
    float* __restrict__ Out, int ntargets, int do_relu)
{
    __shared__ float Cs[16 * CS];
    const int wave = threadIdx.x >> 5;
    const int lane = threadIdx.x & 31;
    const int half = lane >> 4;
    const int l16  = lane & 15;
    const int koff = half * 2;
    const int col  = wave * 16 + l16;
    const float bcol = bias[col];

    const int ntiles = ntargets >> 4;
    for (int tile = blockIdx.x; tile < ntiles; tile += gridDim.x) {
        const int base = tile * 16;
        {
            // thread t: row r = t>>4, column chunk cl = t&15 -> cols [8*cl, 8*cl+8)
            const int r  = threadIdx.x >> 4;
            const int cl = threadIdx.x & 15;
            const int node = base + r;
            // self half (columns 0..127): raw Xself rows gathered by self_idx
            const int si = selfi[node];
            const float* srow = Xself + (size_t)si * DIM + cl * 8;
            *(float4*)&Cs[r * CS + cl * 8]     = *(const float4*)(srow);
            *(float4*)&Cs[r * CS + cl * 8 + 4] = *(const float4*)(srow + 4);
            // agg half (columns 128..255): 25-way max over Hn rows
            const int* nrow = neigh + (size_t)node * SAMP;
            const float* h0 = Hn + (size_t)nrow[0] * DIM + cl * 8;
            float4 m0 = *(const float4*)(h0);
            float4 m1 = *(const float4*)(h0 + 4);
            for (int s = 1; s < SAMP; ++s) {
                const float* hr = Hn + (size_t)nrow[s] * DIM + cl * 8;
                m0 = f4max(m0, *(const float4*)(hr));
                m1 = f4max(m1, *(const float4*)(hr + 4));
            }
            *(float4*)&Cs[r * CS + DIM + cl * 8]     = m0;
            *(float4*)&Cs[r * CS + DIM + cl * 8 + 4] = m1;
        }
        __syncthreads();

        v8f acc = {};
        // K = 256, 64 WMMA steps; preload B in 4 chunks of 16 (32 VGPRs live)
#pragma unroll
        for (int ch = 0; ch < 4; ++ch) {
            v2f br[16];
#pragma unroll
            for (int s = 0; s < 16; ++s)
                br[s] = *(const v2f*)(W + (size_t)col * (2 * DIM) + ch * 64 + s * 4 + koff);
#pragma unroll
            for (int s = 0; s < 16; ++s) {
                v2f av = *(const v2f*)&Cs[l16 * CS + ch * 64 + s * 4 + koff];
                acc = __builtin_amdgcn_wmma_f32_16x16x4_f32(
                    false, av, false, br[s], (short)0, acc, false, false);
            }
        }

        float* dst = Out + (size_t)base * DIM;
#pragma unroll
        for (int r = 0; r < 8; ++r) {
            float v = acc[r] + bcol;
            if (do_relu) v = v > 0.f ? v : 0.f;
            dst[(r + half * 8) * DIM + col] = v;
        }
        __syncthreads();
    }
}

// --------------------------------------------------------------------------
// BatchNorm stats: deterministic two-stage column reduction (no float atomics)
// --------------------------------------------------------------------------
__global__ void __launch_bounds__(128) bn_partial(
    const float* __restrict__ X, int nrows,
    float* __restrict__ psum, float* __restrict__ psumsq)
{
    const int c = threadIdx.x; // 128 columns
    float s = 0.f, sq = 0.f;
    for (int r = blockIdx.x; r < nrows; r += gridDim.x) {
        float v = X[(size_t)r * DIM + c];
        s += v; sq += v * v;
    }
    psum[blockIdx.x * DIM + c]   = s;
    psumsq[blockIdx.x * DIM + c] = sq;
}

__global__ void __launch_bounds__(128) bn_finalize(
    const float* __restrict__ psum, const float* __restrict__ psumsq,
    const float* __restrict__ gamma, const float* __restrict__ beta,
    float* __restrict__ coef, int nrows, int nblocks)
{
    const int c = threadIdx.x;
    float s = 0.f, sq = 0.f;
    for (int b = 0; b < nblocks; ++b) {
        s  += psum[b * DIM + c];
        sq += psumsq[b * DIM + c];
    }
    const float inv_n = 1.f / (float)nrows;
    const float mu  = s * inv_n;
    const float var = sq * inv_n - mu * mu;   // biased, matches jnp.var
    const float istd = rsqrtf(var + 1e-5f);
    const float scale = istd * gamma[c];
    coef[c]        = scale;                    // y = x*scale + shift
    coef[DIM + c]  = beta[c] - mu * scale;
}

// --------------------------------------------------------------------------
// Apply BN (scale/shift) then row-wise L2 normalize: x <- y / (||y||_2 + 1e-6)
// 128 threads per block = 4 wave32; one row per block iteration.
// --------------------------------------------------------------------------
__global__ void __launch_bounds__(128) bn_l2norm(
    float* __restrict__ X, const float* __restrict__ coef, int nrows)
{
    __shared__ float red[4];
    const int c = threadIdx.x;
    const float scale = coef[c];
    const float shift = coef[DIM + c];
    for (int r = blockIdx.x; r < nrows; r += gridDim.x) {
        float y = X[(size_t)r * DIM + c] * scale + shift;
        float sq = y * y;
        // wave32 butterfly reduction
        for (int off = 16; off > 0; off >>= 1)
            sq += __shfl_xor(sq, off, 32);
        if ((threadIdx.x & 31) == 0) red[threadIdx.x >> 5] = sq;
        __syncthreads();
        float tot = red[0] + red[1] + red[2] + red[3];
        X[(size_t)r * DIM + c] = y / (sqrtf(tot) + 1e-6f);
        __syncthreads();
    }
}

// --------------------------------------------------------------------------
extern "C" void kernel_launch(void* const* d_in, const int* in_sizes, int n_in,
                              void* d_out, int out_size, void* d_ws, size_t ws_size,
                              hipStream_t stream) {
    const float* features = (const float*)d_in[0];   // [N0,128]
    const float* Wp0      = (const float*)d_in[1];   // [128,128]
    const float* bp0      = (const float*)d_in[2];   // [128]
    const float* Wp1      = (const float*)d_in[3];   // [128,128]
    const float* bp1      = (const float*)d_in[4];   // [128]
    const float* W0       = (const float*)d_in[5];   // [128,256]
    const float* b0       = (const float*)d_in[6];   // [128]
    const float* W1       = (const float*)d_in[7];   // [128,256]
    const float* b1       = (const float*)d_in[8];   // [128]
    const float* gamma    = (const float*)d_in[9];   // [128]
    const float* beta     = (const float*)d_in[10];  // [128]
    const int* neigh0     = (const int*)d_in[11];    // [N1,25]
    const int* self0      = (const int*)d_in[12];    // [N1]
    const int* neigh1     = (const int*)d_in[13];    // [N2,25]
    const int* self1      = (const int*)d_in[14];    // [N2]
    float* out            = (float*)d_out;           // [N2,128]

    // workspace layout (bytes)
    char* ws = (char*)d_ws;
    float* h      = (float*)(ws);                                        // max(N0,N1)*128
    float* mid    = (float*)(ws + (size_t)N0 * DIM * 4);                 // N1*128
    float* psum   = (float*)(ws + (size_t)N0 * DIM * 4 + (size_t)N1 * DIM * 4);
    float* psumsq = psum + STATS_BLOCKS * DIM;
    float* coef   = psumsq + STATS_BLOCKS * DIM;                         // 256 floats

    // ---- Layer 0 ----
    // h0 = relu(features @ Wp0^T + bp0)        [N0,128]
    gemm_nt_relu<<<2048, 256, 0, stream>>>(features, Wp0, bp0, h, N0);
    // mid = relu([features[self0], max agg(h0)] @ W0^T + b0)   [N1,128]
    sage_fused<<<2048, 256, 0, stream>>>(features, h, neigh0, self0, W0, b0,
                                         mid, N1, /*relu=*/1);
    // batchnorm (batch stats) + L2 row normalize, in place
    bn_partial<<<STATS_BLOCKS, 128, 0, stream>>>(mid, N1, psum, psumsq);
    bn_finalize<<<1, 128, 0, stream>>>(psum, psumsq, gamma, beta, coef, N1, STATS_BLOCKS);
    bn_l2norm<<<2048, 128, 0, stream>>>(mid, coef, N1);

    // ---- Layer 1 ----
    // h1 = relu(mid @ Wp1^T + bp1)             [N1,128]  (reuses h buffer)
    gemm_nt_relu<<<2048, 256, 0, stream>>>(mid, Wp1, bp1, h, N1);
    // out = [mid[self1], max agg(h1)] @ W1^T + b1          [N2,128], no relu
    sage_fused<<<2048, 256, 0, stream>>>(mid, h, neigh1, self1, W1, b1,
                                         out, N2, /*relu=*/0);
}